// CTCPredictionsLayer_79783312490907
// MI455X (gfx1250) — compile-verified
//
#include <hip/hip_runtime.h>
#include <stdint.h>

#define T_   512
#define B_   32
#define C_   2048
#define KB   16
#define NTH  128
#define NEGF (-1e30f)

typedef unsigned int u32;
typedef unsigned long long u64;
typedef __attribute__((ext_vector_type(4))) u32 v4u_t;
typedef __attribute__((ext_vector_type(8))) int v8i_t;
typedef __attribute__((ext_vector_type(4))) int v4i_t;

__device__ inline float lae(float a, float b) {
  float m = fmaxf(a, b);
  if (m <= -1e29f) return NEGF;
  float n = fminf(a, b);
  return m + log1pf(__expf(n - m));
}
// monotonic float -> u32 ordering (larger key == larger float)
__device__ inline u32 ford(float f) {
  u32 u = __float_as_uint(f);
  return (u & 0x80000000u) ? ~u : (u | 0x80000000u);
}
__device__ inline float ford_inv(u32 v) {
  u32 u = (v & 0x80000000u) ? (v & 0x7FFFFFFFu) : ~v;
  return __uint_as_float(u);
}
__device__ inline u64 wave_max64(u64 v) {
#pragma unroll
  for (int off = 16; off; off >>= 1) {
    u64 o = __shfl_xor(v, off, 32);
    if (o > v) v = o;
  }
  return v;
}

// TDM: async-load one C_-float row (2048 x 1 tile, 4B elements) into LDS.
// D# bit layout per CDNA5 ISA ch.8 (group0: count/lds_addr/global_addr/type=2,
// group1: data_size=4B, tensor_dim0=C, tensor_dim1=1, tile=Cx1, stride0=C).
__device__ inline void tdm_load_row(const float* gsrc, u32 lds_byte_off) {
  u64 ga = (u64)(uintptr_t)gsrc;
  v4u_t g0;
  g0.x = 1u;                                            // count=1, user desc
  g0.y = lds_byte_off;                                  // LDS dest byte addr
  g0.z = (u32)ga;                                       // global_addr[31:0]
  g0.w = ((u32)(ga >> 32) & 0x01FFFFFFu) | (2u << 30);  // addr[56:32] | type=2
  v8i_t g1;
  g1[0] = (int)(2u << 16);                    // data_size = 2 (4 bytes)
  g1[1] = (int)(((u32)C_ & 0xFFFFu) << 16);   // tensor_dim0[15:0]
  g1[2] = (int)(((u32)C_ >> 16) | (1u << 16));// tensor_dim0[31:16] | tensor_dim1=1
  g1[3] = (int)(((u32)C_ & 0xFFFFu) << 16);   // tile_dim0 = C
  g1[4] = 1;                                  // tile_dim1 = 1, tile_dim2 = 0
  g1[5] = C_;                                 // tensor_dim0_stride
  g1[6] = 0;
  g1[7] = 0;
  v4i_t gz = {0, 0, 0, 0};
#if defined(__clang_major__) && (__clang_major__ >= 23)
  v8i_t gz8 = {0, 0, 0, 0, 0, 0, 0, 0};
  __builtin_amdgcn_tensor_load_to_lds(g0, g1, gz, gz, gz8, 0);
#else
  __builtin_amdgcn_tensor_load_to_lds(g0, g1, gz, gz, 0);
#endif
}

__global__ __launch_bounds__(NTH) void ctc_beam_kernel(
    const float* __restrict__ data, const int* __restrict__ dlen,
    float* __restrict__ out) {
  __shared__ float lds_lp[2][C_];     // double-buffered log-prob rows (TDM dest)
  __shared__ u32   bp[T_][KB];        // backpointers: bit31=stay, [3:0]=src, [14:4]=cls
  __shared__ u64   red[4];
  __shared__ u64   top17[17];
  __shared__ u64   win[KB];
  __shared__ float s_pb[2][KB], s_pnb[2][KB], s_ptot[KB], s_spb[KB], s_spnb[KB];
  __shared__ int   s_len[2][KB], s_last[2][KB];

  const int tid = threadIdx.x;
  const int b = blockIdx.x;
  int len = dlen[b];
  len = len < 0 ? 0 : (len > T_ ? T_ : len);

  if (tid < KB) {
    s_pb[0][tid]  = (tid == 0) ? 0.0f : NEGF;
    s_pnb[0][tid] = NEGF;
    s_len[0][tid] = 0;
    s_last[0][tid] = -1;
  }
  const u32 lpoff0 = (u32)(uintptr_t)(&lds_lp[0][0]);
  const u32 lpoff1 = (u32)(uintptr_t)(&lds_lp[1][0]);
  if (tid == 0) {  // TDM ignores EXEC; issued once by wave 0
    if (len > 0) tdm_load_row(data + ((size_t)0 * B_ + b) * C_, lpoff0);
    if (len > 1) tdm_load_row(data + ((size_t)1 * B_ + b) * C_, lpoff1);
  }
  __syncthreads();

  int cur = 0;
  for (int t = 0; t < len; ++t) {
    // wave0 issued the TDM ops; it waits, barrier releases the workgroup
    if (tid == 0) {
      if (t + 1 < len) __builtin_amdgcn_s_wait_tensorcnt((short)1);
      else             __builtin_amdgcn_s_wait_tensorcnt((short)0);
    }
    __syncthreads();
    const float* lp = lds_lp[t & 1];

    if (tid < KB) s_ptot[tid] = lae(s_pb[cur][tid], s_pnb[cur][tid]);

    // each thread caches 16 strided lp keys in registers (blank excluded)
    u64 lkey[16];
#pragma unroll
    for (int j = 0; j < 16; ++j) {
      int c = tid + j * NTH;
      lkey[j] = (c == 0) ? 0ull
                         : (((u64)ford(lp[c]) << 16) | (u64)(2047 - c));
    }
    __syncthreads();  // s_ptot visible

    // ---- top-17 of lp (17 argmax rounds) ----
    for (int r = 0; r < 17; ++r) {
      u64 v = 0;
#pragma unroll
      for (int j = 0; j < 16; ++j) if (lkey[j] > v) v = lkey[j];
      v = wave_max64(v);
      if ((tid & 31) == 0) red[tid >> 5] = v;
      __syncthreads();
      u64 m = red[0];
      if (red[1] > m) m = red[1];
      if (red[2] > m) m = red[2];
      if (red[3] > m) m = red[3];
      if (tid == 0) top17[r] = m;
#pragma unroll
      for (int j = 0; j < 16; ++j) if (lkey[j] == m) lkey[j] = 0;
      __syncthreads();
    }

    // ---- build <=304 candidates: 16 stay, 16 special (c==last), 16x17 general
    u64 ckey[3];
#pragma unroll
    for (int q = 0; q < 3; ++q) {
      int i = tid + q * NTH;
      u64 kk = 0;
      if (i < 16) {
        int k = i;
        float pt = s_ptot[k];
        float spb = pt + lp[0];  // stay_pb = p_tot + lp[BLANK]
        int lastk = s_last[cur][k];
        float spnb = (lastk >= 0) ? (s_pnb[cur][k] + lp[lastk]) : NEGF;
        s_spb[k] = spb; s_spnb[k] = spnb;
        float sc = lae(spb, spnb);
        kk = ((u64)ford(sc) << 16) | (u64)(65535 - k);
      } else if (i < 32) {
        int k = i - 16;
        int lastk = s_last[cur][k];
        if (lastk >= 0) {
          float sc = (lastk != 0) ? (lp[lastk] + s_pb[cur][k]) : NEGF;
          u32 idx = 16u + (u32)k * C_ + (u32)lastk;
          kk = ((u64)ford(sc) << 16) | (u64)(65535u - idx);
        }
      } else if (i < 32 + KB * 17) {
        int g = i - 32;
        int k = g / 17, j = g - 17 * k;
        u64 tk = top17[j];
        if (tk != 0) {
          int c = 2047 - (int)(tk & 0xFFFFull);
          if (c != s_last[cur][k]) {
            float sc = ford_inv((u32)(tk >> 16)) + s_ptot[k];
            u32 idx = 16u + (u32)k * C_ + (u32)c;
            kk = ((u64)ford(sc) << 16) | (u64)(65535u - idx);
          }
        }
      }
      ckey[q] = kk;
    }
    __syncthreads();

    // ---- top-16 of candidates (ordered: winner r == new beam r) ----
    for (int r = 0; r < KB; ++r) {
      u64 v = ckey[0];
      if (ckey[1] > v) v = ckey[1];
      if (ckey[2] > v) v = ckey[2];
      v = wave_max64(v);
      if ((tid & 31) == 0) red[tid >> 5] = v;
      __syncthreads();
      u64 m = red[0];
      if (red[1] > m) m = red[1];
      if (red[2] > m) m = red[2];
      if (red[3] > m) m = red[3];
      if (tid == 0) win[r] = m;
      if (ckey[0] == m) ckey[0] = 0;
      if (ckey[1] == m) ckey[1] = 0;
      if (ckey[2] == m) ckey[2] = 0;
      __syncthreads();
    }

    // ---- beam state update (double-buffered) + backpointer ----
    int nxt = cur ^ 1;
    if (tid < KB) {
      u64 w = win[tid];
      u32 idx = 65535u - (u32)(w & 0xFFFFull);
      float npb, npnb; int nlen, nlast; u32 e;
      if (idx < 16u) {
        int s = (int)idx;
        npb = s_spb[s]; npnb = s_spnb[s];
        nlen = s_len[cur][s]; nlast = s_last[cur][s];
        e = 0x80000000u | (u32)s;
      } else {
        u32 x = idx - 16u;
        int s = (int)(x >> 11);
        int cls = (int)(x & 2047u);
        float base = (cls == s_last[cur][s]) ? s_pb[cur][s] : s_ptot[s];
        npnb = lp[cls] + base;
        npb = NEGF;
        nlen = s_len[cur][s] + 1;
        nlast = cls;
        e = (u32)s | ((u32)cls << 4);
      }
      s_pb[nxt][tid] = npb; s_pnb[nxt][tid] = npnb;
      s_len[nxt][tid] = nlen; s_last[nxt][tid] = nlast;
      bp[t][tid] = e;
    }
    __syncthreads();

    // prefetch row t+2 into the buffer just freed (after the barrier above)
    if (tid == 0 && (t + 2) < len) {
      tdm_load_row(data + ((size_t)(t + 2) * B_ + b) * C_,
                   (t & 1) ? lpoff1 : lpoff0);
    }
    cur = nxt;
  }

  // ---- output: pick best of first TOP_PATHS=4 beams, trace back ----
  float* row = out + (size_t)b * T_;
  for (int i = tid; i < T_; i += NTH) row[i] = 0.0f;  // BLANK fill
  __syncthreads();
  if (tid == 0) {
    float best = 3.4e38f; int jb = 0;
    for (int j = 0; j < 4; ++j) {
      float v = -lae(s_pb[cur][j], s_pnb[cur][j]);
      if (v < best) { best = v; jb = j; }
    }
    int c = jb, pos = s_len[cur][jb];
    for (int t = len - 1; t >= 0; --t) {
      u32 e = bp[t][c];
      if (!(e & 0x80000000u)) {
        --pos;
        row[pos] = (float)((e >> 4) & 0x7FFu);
      }
      c = (int)(e & 15u);
    }
  }
}

extern "C" void kernel_launch(void* const* d_in, const int* in_sizes, int n_in,
                              void* d_out, int out_size, void* d_ws, size_t ws_size,
                              hipStream_t stream) {
  const float* data = (const float*)d_in[0];   // [T, B, C] log-probs
  const int* dlen   = (const int*)d_in[1];     // [B]
  float* out        = (float*)d_out;           // [B, T]
  (void)in_sizes; (void)n_in; (void)out_size; (void)d_ws; (void)ws_size;
  hipLaunchKernelGGL(ctc_beam_kernel, dim3(B_), dim3(NTH), 0, stream,
                     data, dlen, out);
}